// SELayer_2559800508872
// MI455X (gfx1250) — compile-verified
//
#include <hip/hip_runtime.h>
#include <math.h>

typedef __attribute__((ext_vector_type(2))) float v2f;
typedef __attribute__((ext_vector_type(4))) float v4f;
typedef __attribute__((ext_vector_type(8))) float v8f;

#define C_DIM 256
#define CRED  16
#define NBATCH 8

// ---------------------------------------------------------------------------
// Kernel 0: zero the accumulation workspace (sums[8*256] + counts[8])
// ---------------------------------------------------------------------------
__global__ void zero_ws_kernel(float* __restrict__ ws, int n) {
  for (int i = blockIdx.x * blockDim.x + threadIdx.x; i < n;
       i += blockDim.x * gridDim.x)
    ws[i] = 0.0f;
}

// ---------------------------------------------------------------------------
// Kernel 1: sorted segment sum.  batch_idx is sorted, so each thread keeps a
// private float4 accumulator + count and flushes with global f32 atomics only
// when the batch id along its strided row sequence changes (<= ~9 flushes).
// Layout: 256 threads = 4 rows x 64 float4 lanes -> 4 KB contiguous per
// block-iteration, fully coalesced b128 loads.
// ---------------------------------------------------------------------------
__device__ __forceinline__ void flush_acc(float* __restrict__ sums,
                                          float* __restrict__ counts, int b,
                                          int c, const v4f acc, float cnt,
                                          bool count_lane) {
  atomicAdd(&sums[b * C_DIM + c + 0], acc.x);
  atomicAdd(&sums[b * C_DIM + c + 1], acc.y);
  atomicAdd(&sums[b * C_DIM + c + 2], acc.z);
  atomicAdd(&sums[b * C_DIM + c + 3], acc.w);
  if (count_lane) atomicAdd(&counts[b], cnt);
}

__global__ void seg_sum_kernel(const float* __restrict__ feat,
                               const int* __restrict__ bidx,
                               float* __restrict__ sums,
                               float* __restrict__ counts, int n_points) {
  const int tid        = threadIdx.x;
  const int row_in_blk = tid >> 6;      // 0..3
  const int c4         = tid & 63;      // float4 lane within the 256-ch row
  const int c          = c4 * 4;
  const int rows_step  = gridDim.x * 4;

  v4f   acc = {0.f, 0.f, 0.f, 0.f};
  float cnt = 0.f;
  int   cur = -1;

  for (int row = blockIdx.x * 4 + row_in_blk; row < n_points; row += rows_step) {
    const int b = bidx[row];
    if (b != cur) {
      if (cur >= 0) flush_acc(sums, counts, cur, c, acc, cnt, c4 == 0);
      cur = b;
      acc = (v4f){0.f, 0.f, 0.f, 0.f};
      cnt = 0.f;
    }
    const v4f f = __builtin_nontemporal_load(
        reinterpret_cast<const v4f*>(feat + (size_t)row * C_DIM + c));
    acc += f;
    cnt += 1.f;
  }
  if (cur >= 0) flush_acc(sums, counts, cur, c, acc, cnt, c4 == 0);
}

// ---------------------------------------------------------------------------
// Kernel 2: gate MLP on the matrix path, one wave32.
//   y[16x256] (rows 8..15 zero-padded)  ->  h = relu(y @ w1)  [16x16]
//   g = sigmoid(h @ w2)                 ->  gate[8x256]
// Uses V_WMMA_F32_16X16X4_F32 (full f32, matches reference precision).
//   A 16x4 layout: lane l: row = l&15; VGPR0/1 hold K = (l<16 ? 0,1 : 2,3)
//   B 4x16 layout: lane l: col = l&15; VGPR0/1 hold K = (l<16 ? 0,1 : 2,3)
//   C 16x16:       VGPR v, lane l: row = v + (l<16 ? 0 : 8), col = l&15
// ---------------------------------------------------------------------------
__global__ void __launch_bounds__(32)
gate_kernel(const float* __restrict__ sums, const float* __restrict__ counts,
            const float* __restrict__ w1, const float* __restrict__ w2,
            float* __restrict__ gate) {
  __shared__ float yl[16 * C_DIM];   // 16 KB: padded mean-pooled features
  __shared__ float hl[16 * 16];      // hidden activations

  const int lane = threadIdx.x;      // 0..31, single wave

  // mean pool: y = sums / max(counts, 1); rows 8..15 = 0
  for (int i = lane; i < 16 * C_DIM; i += 32) {
    const int r = i >> 8;
    float v = 0.f;
    if (r < NBATCH) v = sums[i] * (1.f / fmaxf(counts[r], 1.f));
    yl[i] = v;
  }
  __syncthreads();

  const int row  = lane & 15;
  const int koff = (lane < 16) ? 0 : 2;

  // GEMM1: h = y @ w1  (M=16, N=16, K=256) -> 64 x wmma_f32_16x16x4_f32
  v8f acc = {};
  for (int kk = 0; kk < C_DIM / 4; ++kk) {
    const int k = kk * 4 + koff;
    v2f a, b;
    a.x = yl[row * C_DIM + k];
    a.y = yl[row * C_DIM + k + 1];
    b.x = w1[k * CRED + row];          // w1[k][n], n = lane&15
    b.y = w1[(k + 1) * CRED + row];
    acc = __builtin_amdgcn_wmma_f32_16x16x4_f32(false, a, false, b, (short)0,
                                                acc, false, false);
  }

  // ReLU, scatter C layout -> hl[16][16] (rows 8..15 are exactly zero)
#pragma unroll
  for (int v = 0; v < 8; ++v) {
    const int m = v + ((lane < 16) ? 0 : 8);
    hl[m * 16 + (lane & 15)] = fmaxf(acc[v], 0.f);
  }
  __syncthreads();

  // GEMM2: g = h @ w2  (M=16, K=16, N=256) -> 16 column tiles x 4 wmma each
  for (int j = 0; j < 16; ++j) {
    v8f acc2 = {};
    for (int kk = 0; kk < 4; ++kk) {
      const int k = kk * 4 + koff;
      v2f a, b;
      a.x = hl[row * 16 + k];
      a.y = hl[row * 16 + k + 1];
      b.x = w2[k * C_DIM + j * 16 + row];        // w2[k][j*16 + n]
      b.y = w2[(k + 1) * C_DIM + j * 16 + row];
      acc2 = __builtin_amdgcn_wmma_f32_16x16x4_f32(false, a, false, b,
                                                   (short)0, acc2, false,
                                                   false);
    }
    // sigmoid + store gate rows 0..7 (lanes 0..15 hold rows v = 0..7)
    if (lane < 16) {
#pragma unroll
      for (int v = 0; v < 8; ++v) {
        const float x = acc2[v];
        gate[v * C_DIM + j * 16 + lane] = 1.f / (1.f + __expf(-x));
      }
    }
  }
}

// ---------------------------------------------------------------------------
// Kernel 3: out[p, c] = features[p, c] * gate[batch_idx[p], c]
// Pure streaming: NT b128 loads of features, NT b128 stores of output;
// gate (8 KB) and batch_idx (2 MB) stay cache-resident.
// ---------------------------------------------------------------------------
__global__ void scale_kernel(const float* __restrict__ feat,
                             const int* __restrict__ bidx,
                             const float* __restrict__ gate,
                             float* __restrict__ out, int n_points) {
  const int idx = blockIdx.x * blockDim.x + threadIdx.x;  // one float4 each
  const int row = idx >> 6;
  if (row >= n_points) return;
  const int c = (idx & 63) * 4;
  const int b = bidx[row];

  const v4f f = __builtin_nontemporal_load(
      reinterpret_cast<const v4f*>(feat + (size_t)row * C_DIM + c));
  const v4f g =
      *reinterpret_cast<const v4f*>(gate + b * C_DIM + c);
  __builtin_nontemporal_store(
      f * g, reinterpret_cast<v4f*>(out + (size_t)row * C_DIM + c));
}

// ---------------------------------------------------------------------------
// Host launcher
// ---------------------------------------------------------------------------
extern "C" void kernel_launch(void* const* d_in, const int* in_sizes, int n_in,
                              void* d_out, int out_size, void* d_ws,
                              size_t ws_size, hipStream_t stream) {
  const float* feat = (const float*)d_in[0];   // [N, 256] f32
  const float* w1   = (const float*)d_in[1];   // [256, 16] f32
  const float* w2   = (const float*)d_in[2];   // [16, 256] f32
  const int*   bidx = (const int*)d_in[3];     // [N] i32, sorted
  const int n_points = in_sizes[3];

  float* ws     = (float*)d_ws;
  float* sums   = ws;                 // 8*256 floats
  float* counts = ws + NBATCH * C_DIM;         // 8 floats
  float* gate   = ws + NBATCH * C_DIM + 8;     // 8*256 floats (16B aligned)

  zero_ws_kernel<<<4, 256, 0, stream>>>(ws, NBATCH * C_DIM + 8);

  seg_sum_kernel<<<1024, 256, 0, stream>>>(feat, bidx, sums, counts, n_points);

  gate_kernel<<<1, 32, 0, stream>>>(sums, counts, w1, w2, gate);

  const int total4 = n_points * (C_DIM / 4);   // one float4 per thread
  scale_kernel<<<(total4 + 255) / 256, 256, 0, stream>>>(feat, bidx, gate,
                                                         (float*)d_out,
                                                         n_points);
}